// QuantumLayer_65481071405378
// MI455X (gfx1250) — compile-verified
//
#include <hip/hip_runtime.h>
#include <math.h>

// ---------------------------------------------------------------------------
// 20-qubit statevector simulator for MI455X (gfx1250, wave32).
//  * low wires (7..19)  : fused in LDS (async global->LDS fill/drain),
//                         13 gates per state pass
//  * high wires (0..6)  : fused into a 128x128 unitary, applied as a complex
//                         GEMM with V_WMMA_F32_16X16X4_F32
//  * CNOT ladder        : inverse-Gray permutation out[c]=in[c^(c>>1)],
//                         fused as a gather into the next stage's LDS fill;
//                         the final ladder folds into the expval sign logic.
// ---------------------------------------------------------------------------

#define NQ   20
#define NL   4
#define NST  (NL + 1)               // encoding stage + 4 layers
#define NAMP (1u << NQ)             // 1,048,576 amplitudes
#define HIGHQ 7                     // wires 0..6 -> 128 rows
#define ROWS 128
#define COLS (NAMP / ROWS)          // 8192
#define CHUNK 8192                  // amplitudes per LDS workgroup (64 KB)

typedef float v2f __attribute__((ext_vector_type(2)));
typedef float v8f __attribute__((ext_vector_type(8)));
typedef __attribute__((address_space(3))) float lds_f;   // 32-bit LDS pointer

// -------------------------- gate-table construction ------------------------
// gates layout: [stage][qubit][entry(00,01,10,11)][re,im]  => stage*160+q*8
__global__ void build_gates_kernel(const float* __restrict__ inputs,
                                   const float* __restrict__ weights,
                                   float* __restrict__ gates) {
    int t = blockIdx.x * blockDim.x + threadIdx.x;
    if (t >= NST * NQ) return;
    int s = t / NQ, q = t % NQ;
    float g[8];
    if (s == 0) {                       // RX(inputs[q])
        float th = inputs[q] * 0.5f;
        float c = cosf(th), sn = sinf(th);
        g[0] = c;  g[1] = 0.f;          // g00
        g[2] = 0.f; g[3] = -sn;         // g01 = -i sin
        g[4] = 0.f; g[5] = -sn;         // g10 = -i sin
        g[6] = c;  g[7] = 0.f;          // g11
    } else {                            // Rot(phi,theta,omega) = RZ RY RZ
        int l = s - 1;
        const float* w = weights + (l * NQ + q) * 3;
        float phi = w[0], th = w[1] * 0.5f, om = w[2];
        float c = cosf(th), sn = sinf(th);
        float a = (phi + om) * 0.5f, b = (phi - om) * 0.5f;
        float ca = cosf(a), sa = sinf(a), cb = cosf(b), sb = sinf(b);
        g[0] =  ca * c;  g[1] = -sa * c;    // e^{-ia} c
        g[2] = -cb * sn; g[3] = -sb * sn;   // -e^{ib} s
        g[4] =  cb * sn; g[5] = -sb * sn;   // e^{-ib} s
        g[6] =  ca * c;  g[7] =  sa * c;    // e^{ia} c
    }
    float* out = gates + s * (NQ * 8) + q * 8;
#pragma unroll
    for (int i = 0; i < 8; ++i) out[i] = g[i];
}

// U[s] = u_{w0} (x) ... (x) u_{w6}   (128x128 complex, wires 0..6 = high bits)
// layout: Ubase + s*2*16384 : [re 16384][im 16384], row-major 128x128
__global__ void build_U_kernel(const float* __restrict__ gates,
                               float* __restrict__ Ubase) {
    int t = blockIdx.x * blockDim.x + threadIdx.x;
    if (t >= NST * ROWS * ROWS) return;
    int s = t >> 14;
    int rc = t & 16383;
    int r = rc >> 7, c = rc & 127;
    const float* gs = gates + s * (NQ * 8);
    float pr = 1.f, pi = 0.f;
#pragma unroll
    for (int w = 0; w < HIGHQ; ++w) {       // wire w -> bit (6-w) of r/c
        int br = (r >> (6 - w)) & 1;
        int bc = (c >> (6 - w)) & 1;
        const float* g = gs + w * 8 + (br * 2 + bc) * 2;
        float gr = g[0], gi = g[1];
        float nr = pr * gr - pi * gi;
        float ni = pr * gi + pi * gr;
        pr = nr; pi = ni;
    }
    float* U = Ubase + s * (2 * ROWS * ROWS);
    U[rc]           = pr;
    U[16384 + rc]   = pi;
}

// ------------------------------ state init ---------------------------------
__global__ void init_state_kernel(float* __restrict__ re, float* __restrict__ im) {
    unsigned idx = blockIdx.x * blockDim.x + threadIdx.x;
    re[idx] = (idx == 0u) ? 1.f : 0.f;
    im[idx] = 0.f;
}

// --------------------- low wires (7..19) fused in LDS ----------------------
// Fill LDS with the CDNA5 async global->LDS path (ASYNCcnt-tracked), apply
// 13 gates with barriers, drain with async LDS->global. When doGray != 0 the
// fill gathers src = idx ^ (idx>>1), applying the previous layer's CNOT
// ladder for free (per-lane B32 async loads).
__global__ void __launch_bounds__(256)
low_gates_kernel(const float* __restrict__ inRe, const float* __restrict__ inIm,
                 float* __restrict__ outRe, float* __restrict__ outIm,
                 const float* __restrict__ gstage, int doGray) {
    __shared__ __align__(16) float sre[CHUNK];
    __shared__ __align__(16) float sim[CHUNK];
    unsigned base = blockIdx.x * CHUNK;

    if (doGray) {
        // gathered fill: one 32-bit async load per element per lane
#pragma unroll
        for (int e = 0; e < CHUNK / 256; ++e) {
            unsigned i   = e * 256 + threadIdx.x;
            unsigned idx = base + i;
            unsigned src = idx ^ (idx >> 1);
            asm volatile("global_load_async_to_lds_b32 %0, %1, off"
                         :: "v"((lds_f*)&sre[i]), "v"(inRe + src) : "memory");
            asm volatile("global_load_async_to_lds_b32 %0, %1, off"
                         :: "v"((lds_f*)&sim[i]), "v"(inIm + src) : "memory");
        }
    } else {
        // contiguous fill: 128-bit async loads, fully coalesced
#pragma unroll
        for (int e = 0; e < CHUNK / (256 * 4); ++e) {
            unsigned i = (e * 256 + threadIdx.x) * 4;
            asm volatile("global_load_async_to_lds_b128 %0, %1, off"
                         :: "v"((lds_f*)&sre[i]), "v"(inRe + base + i) : "memory");
            asm volatile("global_load_async_to_lds_b128 %0, %1, off"
                         :: "v"((lds_f*)&sim[i]), "v"(inIm + base + i) : "memory");
        }
    }
    asm volatile("s_wait_asynccnt 0x0" ::: "memory");
    __syncthreads();

    for (int w = HIGHQ; w < NQ; ++w) {           // strides 4096 .. 1
        int sh = 19 - w;
        unsigned st = 1u << sh;
        const float* g = gstage + w * 8;
        float g00r = g[0], g00i = g[1], g01r = g[2], g01i = g[3];
        float g10r = g[4], g10i = g[5], g11r = g[6], g11i = g[7];
#pragma unroll
        for (int e = 0; e < (CHUNK / 2) / 256; ++e) {   // 16 pairs/thread
            unsigned p  = e * 256 + threadIdx.x;
            unsigned i0 = ((p >> sh) << (sh + 1)) | (p & (st - 1u));
            unsigned i1 = i0 + st;
            float a0r = sre[i0], a0i = sim[i0];
            float a1r = sre[i1], a1i = sim[i1];
            sre[i0] = g00r * a0r - g00i * a0i + g01r * a1r - g01i * a1i;
            sim[i0] = g00r * a0i + g00i * a0r + g01r * a1i + g01i * a1r;
            sre[i1] = g10r * a0r - g10i * a0i + g11r * a1r - g11i * a1i;
            sim[i1] = g10r * a0i + g10i * a0r + g11r * a1i + g11i * a1r;
        }
        __syncthreads();                 // also completes DS stores (DScnt)
    }

    // contiguous drain: 128-bit async LDS->global stores
#pragma unroll
    for (int e = 0; e < CHUNK / (256 * 4); ++e) {
        unsigned i = (e * 256 + threadIdx.x) * 4;
        asm volatile("global_store_async_from_lds_b128 %0, %1, off"
                     :: "v"(outRe + base + i), "v"((lds_f*)&sre[i]) : "memory");
        asm volatile("global_store_async_from_lds_b128 %0, %1, off"
                     :: "v"(outIm + base + i), "v"((lds_f*)&sim[i]) : "memory");
    }
    asm volatile("s_wait_asynccnt 0x0" ::: "memory");
}

// ---------- high wires (0..6): complex GEMM  psi' = U(128x128) psi ---------
// One wave computes one 16x16 complex tile with V_WMMA_F32_16X16X4_F32.
// f32 A 16x4 layout : lanes 0-15 => M=lane, K = k+{0,1}; lanes 16-31 => K = k+{2,3}
// f32 B 4x16 layout : v[j] holds row K = k + j + 2*(lane>=16), col = lane&15
// f32 C 16x16      : VGPR r => row (r | 8*(lane>=16)), col = lane&15
__global__ void __launch_bounds__(256)
high_matmul_kernel(const float* __restrict__ Ure, const float* __restrict__ Uim,
                   const float* __restrict__ inRe, const float* __restrict__ inIm,
                   float* __restrict__ outRe, float* __restrict__ outIm) {
    int lane = threadIdx.x & 31;
    int wave = threadIdx.x >> 5;
    int tile = blockIdx.x * 8 + wave;        // 4096 tiles total
    int rowBase = (tile & 7) * 16;           // 8 row tiles
    int colBase = (tile >> 3) * 16;          // 512 col tiles
    int m    = lane & 15;
    int koff = (lane >> 4) * 2;

    v8f cre = {};
    v8f cim = {};

    for (int k = 0; k < ROWS; k += 4) {
        const float* ar = Ure + (rowBase + m) * ROWS + k + koff;
        const float* ai = Uim + (rowBase + m) * ROWS + k + koff;
        v2f Are = { ar[0], ar[1] };
        v2f Aim = { ai[0], ai[1] };
        v2f nAim = -Aim;                     // F32 WMMA: NEG[0:1] must be 0 -> negate in VALU

        int brow = k + koff;
        v2f Bre = { inRe[(unsigned)brow * COLS + colBase + m],
                    inRe[(unsigned)(brow + 1) * COLS + colBase + m] };
        v2f Bim = { inIm[(unsigned)brow * COLS + colBase + m],
                    inIm[(unsigned)(brow + 1) * COLS + colBase + m] };

        // Cre += Are*Bre - Aim*Bim ; Cim += Are*Bim + Aim*Bre
        cre = __builtin_amdgcn_wmma_f32_16x16x4_f32(false, Are,  false, Bre, (short)0, cre, false, false);
        cre = __builtin_amdgcn_wmma_f32_16x16x4_f32(false, nAim, false, Bim, (short)0, cre, false, false);
        cim = __builtin_amdgcn_wmma_f32_16x16x4_f32(false, Are,  false, Bim, (short)0, cim, false, false);
        cim = __builtin_amdgcn_wmma_f32_16x16x4_f32(false, Aim,  false, Bre, (short)0, cim, false, false);
    }

#pragma unroll
    for (int r = 0; r < 8; ++r) {
        unsigned row = rowBase + r + (lane >> 4) * 8;
        unsigned col = colBase + m;
        outRe[row * COLS + col] = cre[r];
        outIm[row * COLS + col] = cim[r];
    }
}

// ------------------------- <Z_q> expectation values ------------------------
// The final CNOT ladder is folded in: logical index c = prefix-XOR (from MSB)
// of the physical index, computed with the 5-step parallel suffix-parity.
__global__ void __launch_bounds__(256)
expval_partial_kernel(const float* __restrict__ re, const float* __restrict__ im,
                      float* __restrict__ partials) {
    __shared__ float red[256];
    float acc[NQ];
#pragma unroll
    for (int q = 0; q < NQ; ++q) acc[q] = 0.f;
#pragma unroll
    for (int e = 0; e < 16; ++e) {
        unsigned idx = blockIdx.x * 4096u + e * 256u + threadIdx.x;
        float r = re[idx], i = im[idx];
        float p = r * r + i * i;
        unsigned c = idx;                  // forward ladder permutation
        c ^= c >> 1; c ^= c >> 2; c ^= c >> 4; c ^= c >> 8; c ^= c >> 16;
#pragma unroll
        for (int q = 0; q < NQ; ++q)
            acc[q] += ((c >> (19 - q)) & 1u) ? -p : p;
    }
    for (int q = 0; q < NQ; ++q) {
        red[threadIdx.x] = acc[q];
        __syncthreads();
        for (int off = 128; off > 0; off >>= 1) {
            if ((int)threadIdx.x < off) red[threadIdx.x] += red[threadIdx.x + off];
            __syncthreads();
        }
        if (threadIdx.x == 0) partials[blockIdx.x * NQ + q] = red[0];
        __syncthreads();
    }
}

__global__ void expval_final_kernel(const float* __restrict__ partials,
                                    float* __restrict__ out) {
    int q = threadIdx.x;
    if (q < NQ) {
        float s = 0.f;
        for (int b = 0; b < 256; ++b) s += partials[b * NQ + q];
        out[q] = s;
    }
}

// ------------------------------- launcher ----------------------------------
extern "C" void kernel_launch(void* const* d_in, const int* in_sizes, int n_in,
                              void* d_out, int out_size, void* d_ws, size_t ws_size,
                              hipStream_t stream) {
    const float* inputs  = (const float*)d_in[0];   // (20,)
    const float* weights = (const float*)d_in[1];   // (4,20,3)

    float* ws = (float*)d_ws;
    const unsigned N = NAMP;
    float* psiA_re = ws;
    float* psiA_im = ws + N;
    float* psiB_re = ws + 2u * N;
    float* psiB_im = ws + 3u * N;
    float* gates   = ws + 4u * N;                       // 5*20*8 = 800 floats
    float* Ubase   = gates + NST * NQ * 8;              // 5*2*16384 floats
    float* partials = Ubase + NST * 2 * ROWS * ROWS;    // 256*20 floats

    build_gates_kernel<<<1, 128, 0, stream>>>(inputs, weights, gates);
    build_U_kernel<<<(NST * ROWS * ROWS + 255) / 256, 256, 0, stream>>>(gates, Ubase);
    init_state_kernel<<<N / 256, 256, 0, stream>>>(psiA_re, psiA_im);

    // Pipeline per stage: low gates (psiA -> psiB, gathering the previous
    // layer's CNOT ladder for stages >= 2), then the WMMA GEMM psiB -> psiA.
    for (int s = 0; s < NST; ++s) {
        int doGray = (s >= 2) ? 1 : 0;   // no ladder precedes stages 0 and 1
        low_gates_kernel<<<N / CHUNK, 256, 0, stream>>>(
            psiA_re, psiA_im, psiB_re, psiB_im, gates + s * (NQ * 8), doGray);
        const float* Ure = Ubase + s * (2 * ROWS * ROWS);
        const float* Uim = Ure + ROWS * ROWS;
        high_matmul_kernel<<<(ROWS / 16) * (COLS / 16) / 8, 256, 0, stream>>>(
            Ure, Uim, psiB_re, psiB_im, psiA_re, psiA_im);
    }

    // Layer-4 ladder folded into the sign logic of the reduction.
    expval_partial_kernel<<<256, 256, 0, stream>>>(psiA_re, psiA_im, partials);
    expval_final_kernel<<<1, 32, 0, stream>>>(partials, (float*)d_out);
}